// BatchHardTripletLoss_30227979829905
// MI455X (gfx1250) — compile-verified
//
#include <hip/hip_runtime.h>
#include <hip/hip_bf16.h>
#include <math.h>

typedef __attribute__((ext_vector_type(2))) float v2f;
typedef __attribute__((ext_vector_type(8))) float v8f;

#define DIM 128
#define MARGIN 0.2f
#define EPS 1e-6f
// D * EPS * EPS = 128e-12
#define DEPS2 1.28e-10f

// ---------------------------------------------------------------------------
// Kernel 1: per-row sums / sums-of-squares for emb1 and emb2 (one wave per
// row, float4 loads, wave32 shuffle reduction), plus init of pos/neg bits.
// ---------------------------------------------------------------------------
__global__ void prep_kernel(const float* __restrict__ e1,
                            const float* __restrict__ e2,
                            float* __restrict__ n1, float* __restrict__ s1,
                            float* __restrict__ n2, float* __restrict__ s2,
                            unsigned* __restrict__ pb, unsigned* __restrict__ nb,
                            int B) {
    int gwave = (blockIdx.x * blockDim.x + threadIdx.x) >> 5;
    int lane  = threadIdx.x & 31;
    if (gwave >= 2 * B) return;

    const float* src;
    float *np, *sp;
    int row;
    bool first = (gwave < B);
    if (first) { row = gwave;      src = e1; np = n1; sp = s1; }
    else       { row = gwave - B;  src = e2; np = n2; sp = s2; }

    float4 a = *(const float4*)(src + (size_t)row * DIM + lane * 4);
    float s = a.x + a.y + a.z + a.w;
    float q = a.x * a.x + a.y * a.y + a.z * a.z + a.w * a.w;
#pragma unroll
    for (int m = 16; m >= 1; m >>= 1) {
        s += __shfl_xor(s, m, 32);
        q += __shfl_xor(q, m, 32);
    }
    if (lane == 0) {
        np[row] = q;
        sp[row] = s;
        if (first) {
            pb[row] = 0u;          // pos-max init: 0.0f (dist >= 0 always)
            nb[row] = 0x7F800000u; // neg-min init: +inf
        }
    }
}

// ---------------------------------------------------------------------------
// Kernel 2: tiled FP32 WMMA GEMM fused with masked per-row max/min.
// Block = 256 threads (8 waves), one 16-row strip per block; each wave sweeps
// 64 of the 512 16-column tiles with V_WMMA_F32_16X16X4_F32.
// Software-pipelined: ping-pong B-fragment register buffers so the loads for
// tile t+8 overlap the WMMA chain of tile t; dual accumulator chains to break
// WMMA->WMMA RAW serialization on the accumulator.
// ---------------------------------------------------------------------------

// Process one 16x16 tile T using fragments in BFCUR while prefetching the
// fragments for tile T+8 into BFNXT (clamped to T when past the end).
#define TILE_STEP(BFCUR, BFNXT, T)                                             \
    do {                                                                       \
        const int tn_ = (T) + 8;                                               \
        const float* bpn_ = e2 +                                               \
            (size_t)((((tn_ < ntiles) ? tn_ : (T)) * 16) + l16) * DIM +        \
            half * 2;                                                          \
        const int col_ = (T) * 16 + l16;                                       \
        const float n2c_ = n2[col_];                                           \
        const float s2c_ = s2[col_];                                           \
        const bool isPos_ = (tgt[col_] == 1);                                  \
        v8f c0_ = {};                                                          \
        v8f c1_ = {};                                                          \
        _Pragma("unroll")                                                      \
        for (int kk = 0; kk < 16; ++kk) {                                      \
            BFNXT[2 * kk] = *(const v2f*)(bpn_ + (2 * kk) * 4);                \
            c0_ = __builtin_amdgcn_wmma_f32_16x16x4_f32(                       \
                false, af[2 * kk], false, BFCUR[2 * kk], (short)0, c0_,        \
                false, false);                                                 \
            BFNXT[2 * kk + 1] = *(const v2f*)(bpn_ + (2 * kk + 1) * 4);        \
            c1_ = __builtin_amdgcn_wmma_f32_16x16x4_f32(                       \
                false, af[2 * kk + 1], false, BFCUR[2 * kk + 1], (short)0,     \
                c1_, false, false);                                            \
        }                                                                      \
        _Pragma("unroll")                                                      \
        for (int v = 0; v < 8; ++v) {                                          \
            float dot_ = c0_[v] + c1_[v];                                      \
            float sq_ = n1r[v] + n2c_ - 2.0f * dot_ +                          \
                        2.0f * EPS * (s1r[v] - s2c_) + DEPS2;                  \
            float d_ = sqrtf(fmaxf(sq_, 0.0f));                                \
            if (isPos_) pm[v] = fmaxf(pm[v], d_);                              \
            else        nm[v] = fminf(nm[v], d_);                              \
        }                                                                      \
    } while (0)

__global__ void __launch_bounds__(256)
batch_hard_kernel(const float* __restrict__ e1, const float* __restrict__ e2,
                  const int* __restrict__ tgt,
                  const float* __restrict__ n1, const float* __restrict__ s1,
                  const float* __restrict__ n2, const float* __restrict__ s2,
                  unsigned* __restrict__ pb, unsigned* __restrict__ nb,
                  int B) {
    __shared__ float lA[16 * 130];  // padded stride: avoid LDS bank conflicts

    const int row0 = blockIdx.x * 16;
    const int tid  = threadIdx.x;

    // Stage the 16x128 A strip into LDS.
    for (int i = tid; i < 16 * DIM; i += 256) {
        int r = i >> 7, k = i & 127;
        lA[r * 130 + k] = e1[(size_t)(row0 + r) * DIM + k];
    }
    __syncthreads();

    const int wave = tid >> 5;
    const int lane = tid & 31;
    const int half = lane >> 4;   // 0: K/M low half, 1: high half per ISA layout
    const int l16  = lane & 15;

    // Preload all 32 A fragments (K-steps of 4). A-matrix 16x4 f32 layout:
    // lanes 0-15 hold {K=0,K=1}, lanes 16-31 hold {K=2,K=3}, M = lane%16.
    v2f af[32];
    const float* arow = &lA[l16 * 130 + half * 2];
#pragma unroll
    for (int kk = 0; kk < 32; ++kk)
        af[kk] = *(const v2f*)(arow + kk * 4);

    // Per-lane row constants: C/D VGPR v holds M = v + 8*(lane>=16).
    float n1r[8], s1r[8], pm[8], nm[8];
#pragma unroll
    for (int v = 0; v < 8; ++v) {
        int row = row0 + v + half * 8;
        n1r[v] = n1[row];
        s1r[v] = s1[row];
        pm[v] = 0.0f;
        nm[v] = INFINITY;
    }

    const int ntiles = B >> 4;  // 512 (tiles per wave: 64, even)

    // Prologue: load the first tile's B fragments.
    v2f bfA[32], bfB[32];
    {
        const float* bp0 = e2 + (size_t)(wave * 16 + l16) * DIM + half * 2;
#pragma unroll
        for (int kk = 0; kk < 32; ++kk)
            bfA[kk] = *(const v2f*)(bp0 + kk * 4);
    }

    // Main pipelined loop: two tiles per iteration, ping-pong buffers.
    for (int t = wave; t < ntiles; t += 16) {
        TILE_STEP(bfA, bfB, t);
        TILE_STEP(bfB, bfA, t + 8);
    }

    // Butterfly-reduce across the 16 lanes that share each row (masks < 16
    // stay within each 16-lane group).
#pragma unroll
    for (int m = 1; m < 16; m <<= 1) {
#pragma unroll
        for (int v = 0; v < 8; ++v) {
            pm[v] = fmaxf(pm[v], __shfl_xor(pm[v], m, 32));
            nm[v] = fminf(nm[v], __shfl_xor(nm[v], m, 32));
        }
    }

    // dist >= 0  =>  uint ordering == float ordering: native u32 atomics.
    if (l16 == 0) {
#pragma unroll
        for (int v = 0; v < 8; ++v) {
            int row = row0 + v + half * 8;
            atomicMax(&pb[row], __float_as_uint(pm[v]));
            atomicMin(&nb[row], __float_as_uint(nm[v]));
        }
    }
}

// ---------------------------------------------------------------------------
// Kernel 3: single-block finalize: mean over anchors of
// relu(pos_max - neg_min + margin).
// ---------------------------------------------------------------------------
__global__ void finalize_kernel(const unsigned* __restrict__ pb,
                                const unsigned* __restrict__ nb,
                                const int* __restrict__ tgt,
                                float* __restrict__ out, int B) {
    __shared__ float sn[256];
    __shared__ float sd[256];
    float num = 0.0f, den = 0.0f;
    for (int r = threadIdx.x; r < B; r += 256) {
        if (tgt[r] == 1) {
            float pmv = __uint_as_float(pb[r]);
            float nmv = __uint_as_float(nb[r]);
            num += fmaxf(pmv - nmv + MARGIN, 0.0f);
            den += 1.0f;
        }
    }
    sn[threadIdx.x] = num;
    sd[threadIdx.x] = den;
    __syncthreads();
    for (int s = 128; s > 0; s >>= 1) {
        if (threadIdx.x < (unsigned)s) {
            sn[threadIdx.x] += sn[threadIdx.x + s];
            sd[threadIdx.x] += sd[threadIdx.x + s];
        }
        __syncthreads();
    }
    if (threadIdx.x == 0) out[0] = sn[0] / sd[0];
}

// ---------------------------------------------------------------------------
extern "C" void kernel_launch(void* const* d_in, const int* in_sizes, int n_in,
                              void* d_out, int out_size, void* d_ws, size_t ws_size,
                              hipStream_t stream) {
    const float* e1  = (const float*)d_in[0];
    const float* e2  = (const float*)d_in[1];
    const int*   tgt = (const int*)d_in[2];
    float* out = (float*)d_out;

    const int B = in_sizes[2];  // 8192

    float* ws = (float*)d_ws;
    float* n1 = ws;
    float* s1 = ws + B;
    float* n2 = ws + 2 * B;
    float* s2 = ws + 3 * B;
    unsigned* pb = (unsigned*)(ws + 4 * B);
    unsigned* nb = (unsigned*)(ws + 5 * B);

    // Kernel 1: 2*B rows, one wave (32 threads) per row, 8 waves per block.
    int prep_blocks = (2 * B * 32 + 255) / 256;
    prep_kernel<<<prep_blocks, 256, 0, stream>>>(e1, e2, n1, s1, n2, s2, pb, nb, B);

    // Kernel 2: one 16-row strip per block.
    batch_hard_kernel<<<B / 16, 256, 0, stream>>>(e1, e2, tgt, n1, s1, n2, s2,
                                                  pb, nb, B);

    // Kernel 3: single-block finalize.
    finalize_kernel<<<1, 256, 0, stream>>>(pb, nb, tgt, out, B);
}